// CompactBilinearPooling_13280038879738
// MI455X (gfx1250) — compile-verified
//
#include <hip/hip_runtime.h>
#include <math.h>

// CompactBilinearPooling on gfx1250:
//   cbp[b,d] = sum_{c1,c2 : (h1[c1]+h2[c2]) mod 8192 == d} s1[c1]*s2[c2]*G[b,c1,c2]
//   G[b] = X_b @ X_b^T,  X_b = x[b] reshaped [512, 196]   (Gram via V_WMMA_F32_16X16X4_F32)
// then signed-sqrt + L2 normalize per batch.

typedef __attribute__((ext_vector_type(2))) float v2f;
typedef __attribute__((ext_vector_type(8))) float v8f;

#define NB 32
#define NC 512
#define HW 196
#define ND 8192

__global__ void cbp_zero_kernel(float* __restrict__ p, int n) {
  int i = blockIdx.x * blockDim.x + threadIdx.x;
  if (i < n) p[i] = 0.0f;
}

// One block per (batch, group-of-8 row-tiles): 32*4 = 128 blocks, 256 threads (8 waves).
// Each wave computes 16x16 Gram tiles with f32 WMMA over K=196 (49 steps of K=4),
// scatters results into a per-block LDS sketch accumulator (ds_add_f32),
// then the block flushes 8192 partial buckets with global f32 atomics.
__global__ __launch_bounds__(256) void cbp_gram_sketch_kernel(
    const float* __restrict__ x, const int* __restrict__ h1,
    const float* __restrict__ s1, const int* __restrict__ h2,
    const float* __restrict__ s2, float* __restrict__ cbp) {
  __shared__ float lds_cbp[ND];  // 32 KB of the 320 KB WGP LDS

  const int b   = blockIdx.x >> 2;  // batch
  const int i8  = blockIdx.x & 3;   // row-tile group: rows [i8*128, i8*128+128)
  const int tid = threadIdx.x;
  const int wave = tid >> 5;        // wave32
  const int lane = tid & 31;
  const int half = lane >> 4;       // ISA A/B f32 layout: lane half selects K pair
  const int lr   = lane & 15;

  for (int idx = tid; idx < ND; idx += 256) lds_cbp[idx] = 0.0f;
  __syncthreads();

  const float* xb = x + (size_t)b * (NC * HW);

  // 8 local row-tiles x 32 col-tiles = 256 tiles per block, 32 per wave.
  for (int t = wave; t < 256; t += 8) {
    const int j  = t & 31;          // col tile (c2)
    const int i  = i8 * 8 + (t >> 5);  // global row tile (c1)

    // A fragment lane addr: row = i*16 + lr, cols k*4 + 2*half + {0,1}
    // B fragment lane addr: row = j*16 + lr, same col pattern (B = X^T tile)
    const float* pa = xb + (i * 16 + lr) * HW + 2 * half;
    const float* pb = xb + (j * 16 + lr) * HW + 2 * half;

    v8f acc = {};
#pragma unroll 7
    for (int k = 0; k < 49; ++k) {   // 196 = 49 * 4
      v2f a  = *(const v2f*)(pa + 4 * k);   // 8B-aligned float2
      v2f bb = *(const v2f*)(pb + 4 * k);
      acc = __builtin_amdgcn_wmma_f32_16x16x4_f32(
          /*neg_a=*/false, a, /*neg_b=*/false, bb,
          /*c_mod=*/(short)0, acc, /*reuse_a=*/false, /*reuse_b=*/false);
    }

    // D layout: VGPR r holds G[m, n], m = r + 8*half (+ tile base), n = lane&15.
    const int   n     = j * 16 + lr;
    const int   h2n   = h2[n];
    const float s2n   = s2[n];
    const int   mbase = i * 16 + half * 8;
#pragma unroll
    for (int r = 0; r < 8; ++r) {
      const int m      = mbase + r;
      const int bucket = (h1[m] + h2n) & (ND - 1);
      atomicAdd(&lds_cbp[bucket], s1[m] * s2n * acc[r]);  // ds_add_f32
    }
  }

  __syncthreads();
  float* dst = cbp + (size_t)b * ND;
  for (int idx = tid; idx < ND; idx += 256) {
    atomicAdd(&dst[idx], lds_cbp[idx]);  // global_atomic_add_f32, 4 blocks/batch
  }
}

// One block per batch: signed sqrt + L2 normalize.
// t = sign(v)*sqrt(|v|+1e-5)  =>  t^2 = |v|+1e-5, so norm needs only one pass.
__global__ __launch_bounds__(256) void cbp_finalize_kernel(
    const float* __restrict__ cbp, float* __restrict__ out) {
  __shared__ float red[256];
  const int b   = blockIdx.x;
  const int tid = threadIdx.x;
  const float* v = cbp + (size_t)b * ND;

  float ssum = 0.0f;
  for (int idx = tid; idx < ND; idx += 256) ssum += fabsf(v[idx]) + 1e-5f;
  red[tid] = ssum;
  __syncthreads();
  for (int s = 128; s > 0; s >>= 1) {
    if (tid < s) red[tid] += red[tid + s];
    __syncthreads();
  }
  const float inv = 1.0f / fmaxf(sqrtf(red[0]), 1e-12f);

  float* o = out + (size_t)b * ND;
  for (int idx = tid; idx < ND; idx += 256) {
    const float val = v[idx];
    const float sg  = (val > 0.0f) ? 1.0f : ((val < 0.0f) ? -1.0f : 0.0f);
    o[idx] = sg * sqrtf(fabsf(val) + 1e-5f) * inv;
  }
}

extern "C" void kernel_launch(void* const* d_in, const int* in_sizes, int n_in,
                              void* d_out, int out_size, void* d_ws, size_t ws_size,
                              hipStream_t stream) {
  const float* x  = (const float*)d_in[0];  // [32, 512, 14, 14] f32
  const int*   h1 = (const int*)d_in[1];    // [512]
  const float* s1 = (const float*)d_in[2];  // [512]
  const int*   h2 = (const int*)d_in[3];    // [512]
  const float* s2 = (const float*)d_in[4];  // [512]
  float*       out = (float*)d_out;         // [32, 8192] f32

  float* cbp = (float*)d_ws;                // 32*8192 f32 = 1 MB scratch
  const int n = NB * ND;

  cbp_zero_kernel<<<(n + 255) / 256, 256, 0, stream>>>(cbp, n);
  cbp_gram_sketch_kernel<<<NB * 4, 256, 0, stream>>>(x, h1, s1, h2, s2, cbp);
  cbp_finalize_kernel<<<NB, 256, 0, stream>>>(cbp, out);
}